// KNN_21526376088017
// MI455X (gfx1250) — compile-verified
//
#include <hip/hip_runtime.h>
#include <hip/hip_bf16.h>

typedef __attribute__((ext_vector_type(2))) float v2f;
typedef __attribute__((ext_vector_type(8))) float v8f;

#define N_TRAIN 50000
#define DIM     64
#define NB      128
#define NCLS    10
#define KSEL    16
#define THREADS 256
#define NWAVES  (THREADS / 32)

static __device__ __forceinline__ unsigned long long pack_key(float d, int idx) {
    return ((unsigned long long)__float_as_uint(d) << 32) | (unsigned int)idx;
}

__global__ __launch_bounds__(THREADS)
void knn_l1_topk_vote(const float* __restrict__ train,
                      const float* __restrict__ target,
                      const float* __restrict__ xtest,
                      float* __restrict__ out) {
    __shared__ float xs[DIM];
    __shared__ unsigned long long cand[THREADS * KSEL];   // 32 KB
    __shared__ unsigned long long red[THREADS];
    __shared__ int winIdx[KSEL];
    __shared__ float votes[NCLS];

    const int tid  = threadIdx.x;
    const int b    = blockIdx.x;
    const int lane = tid & 31;
    const int wave = tid >> 5;

    if (tid < DIM) xs[tid] = xtest[b * DIM + tid];
    __syncthreads();

    // per-lane top-K as sorted packed keys, keys[0] smallest
    unsigned long long keys[KSEL];
#pragma unroll
    for (int j = 0; j < KSEL; ++j) keys[j] = ~0ull;

    v2f ones; ones[0] = 1.0f; ones[1] = 1.0f;

    const int kh = lane >> 4;                 // which K-pair half this lane feeds
    const int m  = lane & 15;                 // A-matrix row (train point within tile)
    const int srcLane = (lane & 8) ? 16 : 0;  // where this row's sum lives in D layout
    const int numPairTiles = (N_TRAIN + 31) / 32;   // 1563

    for (int p = wave; p < numPairTiles; p += NWAVES) {
        const int n0 = p * 32;
        const bool t1ok = (n0 + 16) < N_TRAIN;
        const int row0 = n0 + m;
        const int row1 = t1ok ? (n0 + 16 + m) : (N_TRAIN - 1);
        const float* __restrict__ tp0 = train + (size_t)row0 * DIM + 2 * kh;
        const float* __restrict__ tp1 = train + (size_t)row1 * DIM + 2 * kh;

        // prefetch this wave's next pair-tile (speculative, dropped if OOB-translated)
        {
            int nn = n0 + NWAVES * 32;
            if (nn + m < N_TRAIN)
                __builtin_prefetch(train + (size_t)(nn + m) * DIM, 0, 0);
        }

        v8f acc0 = {};
        v8f acc1 = {};
#pragma unroll
        for (int db = 0; db < DIM; db += 4) {
            v2f xv = *(const v2f*)(xs + db + 2 * kh);
            v2f t0 = *(const v2f*)(tp0 + db);
            v2f t1 = *(const v2f*)(tp1 + db);
            v2f a0, a1;
            a0[0] = __builtin_fabsf(xv[0] - t0[0]);
            a0[1] = __builtin_fabsf(xv[1] - t0[1]);
            a1[0] = __builtin_fabsf(xv[0] - t1[0]);
            a1[1] = __builtin_fabsf(xv[1] - t1[1]);
            // D[m,n] = sum_k A[m,k] * 1  -> every column holds the row's L1 partial sum
            acc0 = __builtin_amdgcn_wmma_f32_16x16x4_f32(false, a0, false, ones,
                                                         (short)0, acc0, false, false);
            acc1 = __builtin_amdgcn_wmma_f32_16x16x4_f32(false, a1, false, ones,
                                                         (short)0, acc1, false, false);
        }

        // Extract per-train-point distance. D layout: row r -> VGPR (r&7),
        // lanes 0-15 if r<8 else lanes 16-31. Lanes 0-15 take tile0, 16-31 tile1.
        float dist = 0.0f;
#pragma unroll
        for (int j = 0; j < 8; ++j) {
            float s0 = __shfl(acc0[j], srcLane, 32);
            float s1 = __shfl(acc1[j], srcLane, 32);
            float pick = (lane & 16) ? s1 : s0;
            if ((lane & 7) == j) dist = pick;
        }

        unsigned long long key = pack_key(dist, n0 + lane);
        if ((lane & 16) && !t1ok) key = ~0ull;   // clamped tail tile: never a candidate

        if (key < keys[KSEL - 1]) {
            keys[KSEL - 1] = key;
#pragma unroll
            for (int j = KSEL - 1; j > 0; --j) {
                unsigned long long x = keys[j - 1], y = keys[j];
                keys[j - 1] = x < y ? x : y;
                keys[j]     = x < y ? y : x;
            }
        }
    }

    // dump sorted per-thread lists to LDS
#pragma unroll
    for (int j = 0; j < KSEL; ++j) cand[tid * KSEL + j] = keys[j];
    __syncthreads();

    // 256-way merge: extract the 16 globally smallest keys
    int myPtr = 0;
    for (int it = 0; it < KSEL; ++it) {
        unsigned long long my = (myPtr < KSEL) ? cand[tid * KSEL + myPtr] : ~0ull;
        red[tid] = my;
        __syncthreads();
#pragma unroll
        for (int s = THREADS / 2; s > 0; s >>= 1) {
            if (tid < s) {
                unsigned long long o = red[tid + s];
                if (o < red[tid]) red[tid] = o;
            }
            __syncthreads();
        }
        unsigned long long w = red[0];
        if (tid == 0) winIdx[it] = (int)(unsigned int)(w & 0xFFFFFFFFu);
        __syncthreads();          // everyone has read red[0] before next round writes
        if (my == w) myPtr++;     // keys are unique (idx in low bits)
    }
    __syncthreads();

    // vote: sum one-hot rows of the 16 winners, then argmax (first max wins)
    if (tid < NCLS) {
        float v = 0.0f;
#pragma unroll
        for (int j = 0; j < KSEL; ++j)
            v += target[(size_t)winIdx[j] * NCLS + tid];
        votes[tid] = v;
    }
    __syncthreads();
    if (tid == 0) {
        int best = 0;
        float bv = votes[0];
#pragma unroll
        for (int c = 1; c < NCLS; ++c) {
            if (votes[c] > bv) { bv = votes[c]; best = c; }
        }
        out[b] = (float)best;
    }
}

extern "C" void kernel_launch(void* const* d_in, const int* in_sizes, int n_in,
                              void* d_out, int out_size, void* d_ws, size_t ws_size,
                              hipStream_t stream) {
    const float* train  = (const float*)d_in[0];   // [50000, 64] f32
    const float* target = (const float*)d_in[1];   // [50000, 10] f32 one-hot
    const float* xtest  = (const float*)d_in[2];   // [128, 64]   f32
    (void)d_in[3];                                  // k == 16, compile-time constant here
    (void)in_sizes; (void)n_in; (void)d_ws; (void)ws_size; (void)out_size;

    knn_l1_topk_vote<<<NB, THREADS, 0, stream>>>(train, target, xtest, (float*)d_out);
}